// OrdRec_35296041239090
// MI455X (gfx1250) — compile-verified
//
#include <hip/hip_runtime.h>
#include <math.h>

typedef float v2f __attribute__((ext_vector_type(2)));
typedef float v4f __attribute__((ext_vector_type(4)));
typedef float v8f __attribute__((ext_vector_type(8)));

#define HIDDEN    512
#define NUM_BINS  10
#define MIN_R     0.5f
#define BIN_SZ    0.5f

// One wave32 computes 16 (user,item) dot products with V_WMMA_F32_16X16X4_F32,
// then lanes 0..15 each finish one pair's ordinal head.
__global__ __launch_bounds__(256) void ordrec_wmma_kernel(
    const int*   __restrict__ uid_input,
    const int*   __restrict__ iid_input,
    const float* __restrict__ uid_features,
    const float* __restrict__ iid_features,
    const float* __restrict__ uid_bias,
    const float* __restrict__ iid_bias,
    const float* __restrict__ uid_t1,
    const float* __restrict__ uid_beta,
    float* __restrict__ out_mass,    // [B,10]
    float* __restrict__ out_mean,    // [B]
    float* __restrict__ out_mode,    // [B]
    float* __restrict__ out_edges,   // [B,10]
    int B)
{
    const int lane        = threadIdx.x & 31;
    const int waveInBlock = threadIdx.x >> 5;
    const int waveGlobal  = blockIdx.x * (blockDim.x >> 5) + waveInBlock;
    const int pairBase    = waveGlobal * 16;
    if (pairBase >= B) return;                 // wave-uniform: EXEC stays all-ones

    const int rowInTile = lane & 15;           // which of the wave's 16 pairs this lane loads
    int pair  = pairBase + rowInTile;
    int pairC = (pair < B) ? pair : (B - 1);   // clamp for ragged tail (loads only)

    const int uid = uid_input[pairC];
    const int iid = iid_input[pairC];

    const float* __restrict__ uRow = uid_features + (size_t)uid * HIDDEN;
    const float* __restrict__ vRow = iid_features + (size_t)iid * HIDDEN;

    // WMMA F32 16x16x4 per-lane layout (both A[M=rowInTile][k] and B[k][N=rowInTile]):
    // lane L holds elements (k + 2*(L>>4)) and (k + 2*(L>>4) + 1)  -> one float2 load each.
    const int colOff = (lane >> 4) << 1;       // 0 for lanes 0-15, 2 for lanes 16-31

    v8f c0 = {0.f,0.f,0.f,0.f,0.f,0.f,0.f,0.f};
    v8f c1 = {0.f,0.f,0.f,0.f,0.f,0.f,0.f,0.f};
    #pragma unroll 4
    for (int k = 0; k < HIDDEN; k += 8) {
        // user rows: ~single-use stream over a 1 GB table -> non-temporal (spare L2)
        v2f a0 = __builtin_nontemporal_load((const v2f*)(uRow + k + colOff));
        v2f a1 = __builtin_nontemporal_load((const v2f*)(uRow + k + 4 + colOff));
        // item rows: ~190 MB unique footprint vs 192 MB L2 -> keep regular-temporal
        v2f b0 = *(const v2f*)(vRow + k + colOff);
        v2f b1 = *(const v2f*)(vRow + k + 4 + colOff);
        // 8 args: (neg_a, A, neg_b, B, c_mod, C, reuse_a, reuse_b)
        c0 = __builtin_amdgcn_wmma_f32_16x16x4_f32(false, a0, false, b0, (short)0, c0, false, false);
        c1 = __builtin_amdgcn_wmma_f32_16x16x4_f32(false, a1, false, b1, (short)0, c1, false, false);
    }
    v8f c = c0 + c1;

    // Diagonal of C: lanes 0-7 hold diag[m]=c[m] (m=lane); lanes 24-31 hold
    // diag[m]=c[lane-24] (m=lane-16). Select then shuffle so lane p holds dot[p].
    int sel = (lane < 16) ? lane : (lane - 24);
    float d = c[0];
    if (sel == 1) d = c[1];
    if (sel == 2) d = c[2];
    if (sel == 3) d = c[3];
    if (sel == 4) d = c[4];
    if (sel == 5) d = c[5];
    if (sel == 6) d = c[6];
    if (sel == 7) d = c[7];
    int srcLane = (rowInTile < 8) ? rowInTile : (rowInTile + 16);
    float dot = __shfl(d, srcLane, 32);

    // Ordinal head: lanes 0..15, one pair each.
    if (lane < 16 && pair < B) {
        const float ub  = uid_bias[uid];
        const float ib  = iid_bias[iid];
        const float t1  = uid_t1[uid];
        const float* __restrict__ bp = uid_beta + (size_t)uid * (NUM_BINS - 2);
        v4f beta_lo = *(const v4f*)(bp);       // 32 B-aligned rows
        v4f beta_hi = *(const v4f*)(bp + 4);
        float beta[NUM_BINS - 2] = { beta_lo.x, beta_lo.y, beta_lo.z, beta_lo.w,
                                     beta_hi.x, beta_hi.y, beta_hi.z, beta_hi.w };

        const float y = dot + ib + ub;

        // Thresholds: T[0]=t1, T[j]=T[j-1]+exp(beta[j-1])
        float T[NUM_BINS - 1];
        T[0] = t1;
        #pragma unroll
        for (int j = 0; j < NUM_BINS - 2; ++j)
            T[j + 1] = T[j] + expf(beta[j]);

        // sigmoid(T - y) = 1/(1+exp(y-T))
        float sig[NUM_BINS - 1];
        #pragma unroll
        for (int k = 0; k < NUM_BINS - 1; ++k)
            sig[k] = 1.0f / (1.0f + expf(y - T[k]));

        float mass[NUM_BINS];
        mass[0] = sig[0];
        #pragma unroll
        for (int k = 1; k < NUM_BINS - 1; ++k)
            mass[k] = sig[k] - sig[k - 1];
        mass[NUM_BINS - 1] = 1.0f - sig[NUM_BINS - 2];

        float mean = 0.0f;
        #pragma unroll
        for (int k = 0; k < NUM_BINS; ++k)
            mean += mass[k] * (MIN_R + BIN_SZ * (float)k);

        // first-occurrence argmax (matches jnp.argmax)
        float best = mass[0]; int bi = 0;
        #pragma unroll
        for (int k = 1; k < NUM_BINS; ++k)
            if (mass[k] > best) { best = mass[k]; bi = k; }
        float mode = MIN_R + BIN_SZ * (float)bi;

        // Stores: rows are 40 B apart -> 8 B aligned, use float2 stores.
        float* mp = out_mass  + (size_t)pair * NUM_BINS;
        float* ep = out_edges + (size_t)pair * NUM_BINS;
        #pragma unroll
        for (int k = 0; k < NUM_BINS; k += 2) {
            v2f m2 = { mass[k], mass[k + 1] };
            *(v2f*)(mp + k) = m2;
            float e0 = (k     < NUM_BINS - 1) ? T[k]     : __builtin_inff();
            float e1 = (k + 1 < NUM_BINS - 1) ? T[k + 1] : __builtin_inff();
            v2f e2 = { e0, e1 };
            *(v2f*)(ep + k) = e2;
        }
        out_mean[pair] = mean;
        out_mode[pair] = mode;
    }
}

extern "C" void kernel_launch(void* const* d_in, const int* in_sizes, int n_in,
                              void* d_out, int out_size, void* d_ws, size_t ws_size,
                              hipStream_t stream) {
    const int*   uid_input    = (const int*)  d_in[0];
    const int*   iid_input    = (const int*)  d_in[1];
    const float* uid_features = (const float*)d_in[2];
    const float* iid_features = (const float*)d_in[3];
    const float* uid_bias     = (const float*)d_in[4];
    const float* iid_bias     = (const float*)d_in[5];
    const float* uid_t1       = (const float*)d_in[6];
    const float* uid_beta     = (const float*)d_in[7];

    const int B = in_sizes[0];
    float* out       = (float*)d_out;
    float* out_mass  = out;                       // B*10
    float* out_mean  = out + (size_t)B * 10;      // B
    float* out_mode  = out + (size_t)B * 11;      // B
    float* out_edges = out + (size_t)B * 12;      // B*10

    const int pairsPerBlock = 128;                // 8 waves * 16 pairs
    const int grid = (B + pairsPerBlock - 1) / pairsPerBlock;
    ordrec_wmma_kernel<<<grid, 256, 0, stream>>>(
        uid_input, iid_input, uid_features, iid_features,
        uid_bias, iid_bias, uid_t1, uid_beta,
        out_mass, out_mean, out_mode, out_edges, B);
}